// ROIBoxHead_78786880078191
// MI455X (gfx1250) — compile-verified
//
#include <hip/hip_runtime.h>
#include <stdint.h>

#define BATCH   2
#define NPROP   1024
#define NCLS    81
#define NFG     80
#define NDET    100
#define IMG_WF  1024.0f
#define IMG_HF  1024.0f
#define SCORE_TH 0.05f
#define NMS_TH  0.5f
#define NEGV    (-1e30f)
#define NEGH    (-5e29f)
#define BBOX_CLIP 4.135166556742356f   /* log(1000/16) */

typedef __attribute__((ext_vector_type(2))) float        v2f;
typedef __attribute__((ext_vector_type(8))) float        v8f;
typedef __attribute__((ext_vector_type(4))) unsigned int v4u;
typedef __attribute__((ext_vector_type(4))) int          v4i;
typedef __attribute__((ext_vector_type(8))) int          v8i;

// ---------------------------------------------------------------------------
// TDM: 1-D tile load (nElem 4-byte elements) global -> LDS.  D# packed per
// CDNA5 ISA (group0: count/lds/global/type, group1:
// data_size=4B, tensor_dim0=nElem, tensor_dim1=1, tile_dim0=nElem, tile_dim1=1).
// 6-arg builtin form (clang-23 / therock-10.0 headers).
// ---------------------------------------------------------------------------
__device__ __forceinline__ void tdm_load_1d(uint32_t ldsAddr, uint64_t gaddr,
                                            uint32_t nElem) {
  v4u g0;
  g0[0] = 1u;                                              // count = 1
  g0[1] = ldsAddr;                                         // lds_addr
  g0[2] = (uint32_t)gaddr;                                 // global_addr[31:0]
  g0[3] = (uint32_t)((gaddr >> 32) & 0x01FFFFFFu) | (2u << 30); // [56:32] | type=2
  v8i g1;
  g1[0] = (int)(2u << 16);                                 // data_size = 4 bytes
  g1[1] = (int)((nElem & 0xFFFFu) << 16);                  // tensor_dim0[15:0]
  g1[2] = (int)(((nElem >> 16) & 0xFFFFu) | (1u << 16));   // td0[31:16] | tensor_dim1=1
  g1[3] = (int)((nElem & 0xFFFFu) << 16);                  // td1 hi=0 | tile_dim0
  g1[4] = 1;                                               // tile_dim1=1, tile_dim2=0
  g1[5] = (int)nElem;                                      // tensor_dim0_stride[31:0]
  g1[6] = 0;
  g1[7] = 0;
  v4i gz4 = {0, 0, 0, 0};
  v8i gz8 = {0, 0, 0, 0, 0, 0, 0, 0};
  __builtin_amdgcn_tensor_load_to_lds(g0, g1, gz4, gz4, gz8, 0);
}

// ---------------------------------------------------------------------------
// Kernel 1: softmax (WMMA row-sum) + per-class box decode + score threshold.
// 256 threads = 8 waves; each wave owns 16 proposals.
// ---------------------------------------------------------------------------
__global__ __launch_bounds__(256) void k1_decode_score(
    const float* __restrict__ logits,   // [B*N, C]
    const float* __restrict__ reg,      // [B*N, C*4]
    const float* __restrict__ props,    // [B*N, 4]
    float* __restrict__ scOut,          // [B*80, N]
    float* __restrict__ bxOut) {        // [B*80, N, 4]
  __shared__ float ssum[128];
  const int lane = threadIdx.x & 31;
  const int wave = threadIdx.x >> 5;
  const int rowBase = blockIdx.x * 128 + wave * 16;

  // ---- softmax denominators for 16 rows via V_WMMA_F32_16X16X4_F32 ----
  // A layout (16x4 f32): lanes 0-15 hold M=lane,K={0,1}; lanes 16-31 K={2,3}.
  const int m = lane & 15;
  const int row = rowBase + m;
  const float* lrow = logits + (size_t)row * NCLS;
  const int khalf = (lane >> 4) << 1;          // 0 or 2
  v8f acc = {0.f, 0.f, 0.f, 0.f, 0.f, 0.f, 0.f, 0.f};
  v2f bone;
  bone.x = 1.0f;
  bone.y = 1.0f;
#pragma unroll
  for (int kc = 0; kc < 21; ++kc) {            // 21*4 = 84 >= 81 classes
    int k0 = kc * 4 + khalf;
    float e0 = (k0 < NCLS) ? __expf(lrow[k0]) : 0.0f;
    float e1 = (k0 + 1 < NCLS) ? __expf(lrow[k0 + 1]) : 0.0f;
    v2f a;
    a.x = e0;
    a.y = e1;
    acc = __builtin_amdgcn_wmma_f32_16x16x4_f32(false, a, false, bone,
                                                (short)0, acc, false, false);
  }
  // D layout: lane 0 VGPR v = rowsum(row v); lane 16 VGPR v = rowsum(row v+8)
  if (lane == 0) {
#pragma unroll
    for (int v = 0; v < 8; ++v) ssum[wave * 16 + v] = acc[v];
  }
  if (lane == 16) {
#pragma unroll
    for (int v = 0; v < 8; ++v) ssum[wave * 16 + 8 + v] = acc[v];
  }
  __syncthreads();

  // ---- 16 rows x 80 fg classes = 1280 (row,class) pairs per wave ----
#pragma unroll 1
  for (int t = 0; t < 40; ++t) {
    int pairIdx = t * 32 + lane;               // 0..1279, class-major
    int rowL = pairIdx & 15;
    int cls = (pairIdx >> 4) + 1;              // 1..80
    int r = rowBase + rowL;
    float denom = ssum[wave * 16 + rowL];
    float scv = __expf(logits[(size_t)r * NCLS + cls]) / denom;

    const float* pr = props + (size_t)r * 4;
    float px1 = pr[0], py1 = pr[1], px2 = pr[2], py2 = pr[3];
    float wdt = px2 - px1 + 1.0f;
    float hgt = py2 - py1 + 1.0f;
    float cx = px1 + 0.5f * wdt;
    float cy = py1 + 0.5f * hgt;
    const float* rg = reg + (size_t)r * (NCLS * 4) + (size_t)cls * 4;
    float dx = rg[0] * 0.1f;                   // / WX
    float dy = rg[1] * 0.1f;                   // / WY
    float dw = fminf(rg[2] * 0.2f, BBOX_CLIP); // / WW, clip
    float dh = fminf(rg[3] * 0.2f, BBOX_CLIP); // / WH, clip
    float pcx = dx * wdt + cx;
    float pcy = dy * hgt + cy;
    float pw = __expf(dw) * wdt;
    float ph = __expf(dh) * hgt;
    float x1 = fminf(fmaxf(pcx - 0.5f * pw, 0.0f), IMG_WF - 1.0f);
    float y1 = fminf(fmaxf(pcy - 0.5f * ph, 0.0f), IMG_HF - 1.0f);
    float x2 = fminf(fmaxf(pcx + 0.5f * pw - 1.0f, 0.0f), IMG_WF - 1.0f);
    float y2 = fminf(fmaxf(pcy + 0.5f * ph - 1.0f, 0.0f), IMG_HF - 1.0f);

    int b = r >> 10;                            // / NPROP
    int n = r & (NPROP - 1);
    size_t o = ((size_t)(b * NFG + (cls - 1)) * NPROP) + n;
    scOut[o] = (scv > SCORE_TH) ? scv : NEGV;
    float4 bb = make_float4(x1, y1, x2, y2);
    *(float4*)(bxOut + o * 4) = bb;
  }
}

// ---------------------------------------------------------------------------
// Kernel 2: per-(image,class) NMS.  160 blocks x 256 threads.
// TDM stages scores/boxes into LDS; bitonic sort; greedy suppression.
// ---------------------------------------------------------------------------
__global__ __launch_bounds__(256) void k2_nms(float* __restrict__ sc,       // [B*80, N] in/out
                                              const float* __restrict__ bx) // [B*80, N, 4]
{
  __shared__ float sScore[NPROP];
  __shared__ int sIdx[NPROP];
  __shared__ float sBox[NPROP * 4];
  __shared__ unsigned char sKeep[NPROP];

  const int tid = threadIdx.x;
  const int slice = blockIdx.x;                // b*80 + c

  for (int i = tid; i < NPROP; i += 256) sIdx[i] = i;

  if (tid < 32) {                              // single wave issues the DMA
    uint64_t gS = (uint64_t)(uintptr_t)(sc + (size_t)slice * NPROP);
    uint64_t gB = (uint64_t)(uintptr_t)(bx + (size_t)slice * NPROP * 4);
    uint32_t ldsS = (uint32_t)(uintptr_t)(&sScore[0]);
    uint32_t ldsB = (uint32_t)(uintptr_t)(&sBox[0]);
    tdm_load_1d(ldsS, gS, NPROP);
    tdm_load_1d(ldsB, gB, NPROP * 4);
    __builtin_amdgcn_s_wait_tensorcnt(0);
  }
  __syncthreads();

  // ---- bitonic sort, descending by score (tie: lower original index) ----
  for (int k = 2; k <= NPROP; k <<= 1) {
    for (int j = k >> 1; j > 0; j >>= 1) {
      __syncthreads();
      for (int r = 0; r < NPROP / 256; ++r) {
        int i = tid + r * 256;
        int l = i ^ j;
        if (l > i) {
          float si = sScore[i], sl = sScore[l];
          int ii = sIdx[i], il = sIdx[l];
          bool iBeforeL = (si > sl) || (si == sl && ii < il);
          bool up = ((i & k) == 0);
          bool doSwap = up ? !iBeforeL : iBeforeL;
          if (doSwap) {
            sScore[i] = sl;
            sScore[l] = si;
            sIdx[i] = il;
            sIdx[l] = ii;
          }
        }
      }
    }
  }
  __syncthreads();

  for (int i = tid; i < NPROP; i += 256) sKeep[i] = (sScore[i] > NEGH) ? 1 : 0;
  __syncthreads();

  // ---- greedy suppression in sorted order (early-out on invalid tail) ----
  for (int i = 0; i < NPROP; ++i) {
    float si = sScore[i];                      // uniform LDS broadcast
    if (si <= NEGH) break;
    if (sKeep[i]) {
      int oi = sIdx[i];
      float ax1 = sBox[oi * 4 + 0], ay1 = sBox[oi * 4 + 1];
      float ax2 = sBox[oi * 4 + 2], ay2 = sBox[oi * 4 + 3];
      float areaA = (ax2 - ax1 + 1.0f) * (ay2 - ay1 + 1.0f);
#pragma unroll
      for (int r = 0; r < NPROP / 256; ++r) {
        int jj = tid + r * 256;
        if (jj > i && sKeep[jj]) {
          int oj = sIdx[jj];
          float bx1 = sBox[oj * 4 + 0], by1 = sBox[oj * 4 + 1];
          float bx2 = sBox[oj * 4 + 2], by2 = sBox[oj * 4 + 3];
          float iw = fmaxf(fminf(ax2, bx2) - fmaxf(ax1, bx1) + 1.0f, 0.0f);
          float ih = fmaxf(fminf(ay2, by2) - fmaxf(ay1, by1) + 1.0f, 0.0f);
          float inter = iw * ih;
          float areaB = (bx2 - bx1 + 1.0f) * (by2 - by1 + 1.0f);
          float iou = inter / (areaA + areaB - inter);
          if (iou > NMS_TH) sKeep[jj] = 0;
        }
      }
    }
    __syncthreads();
  }
  __syncthreads();

  for (int i = tid; i < NPROP; i += 256) {
    int oi = sIdx[i];
    sc[(size_t)slice * NPROP + oi] = sKeep[i] ? sScore[i] : NEGV;
  }
}

// ---------------------------------------------------------------------------
// Kernel 3: per-image top-100 over 80*1024 survivors; write outputs.
// out layout: boxes[B*D*4] | scores[B*D] | labels[B*D] (labels as float).
// ---------------------------------------------------------------------------
__global__ __launch_bounds__(1024) void k3_topd(float* __restrict__ sc,      // [B, 80*N] consumed
                                                const float* __restrict__ bx,
                                                float* __restrict__ out) {
  __shared__ float rv[1024];
  __shared__ int ri[1024];
  const int tid = threadIdx.x;
  const int b = blockIdx.x;
  float* scb = sc + (size_t)b * (NFG * NPROP);

  for (int d = 0; d < NDET; ++d) {
    float best = -3e38f;
    int bidx = 0;
#pragma unroll 1
    for (int q = 0; q < NFG; ++q) {
      int idx = tid + q * 1024;
      float v = scb[idx];
      if (v > best) { best = v; bidx = idx; }  // strict > keeps lowest index
    }
    rv[tid] = best;
    ri[tid] = bidx;
    __syncthreads();
    for (int s = 512; s > 0; s >>= 1) {
      if (tid < s) {
        float v2 = rv[tid + s], v1 = rv[tid];
        int i2 = ri[tid + s], i1 = ri[tid];
        if (v2 > v1 || (v2 == v1 && i2 < i1)) {
          rv[tid] = v2;
          ri[tid] = i2;
        }
      }
      __syncthreads();
    }
    if (tid == 0) {
      float wv = rv[0];
      int wi = ri[0];
      bool ok = wv > NEGH;
      int cls = wi >> 10;                      // / NPROP
      int n = wi & (NPROP - 1);
      float4 bb = make_float4(0.f, 0.f, 0.f, 0.f);
      if (ok) bb = *(const float4*)(bx + ((size_t)(b * NFG + cls) * NPROP + n) * 4);
      size_t bd = (size_t)b * NDET + d;
      out[bd * 4 + 0] = bb.x;
      out[bd * 4 + 1] = bb.y;
      out[bd * 4 + 2] = bb.z;
      out[bd * 4 + 3] = bb.w;
      out[(size_t)BATCH * NDET * 4 + bd] = ok ? wv : 0.0f;
      out[(size_t)BATCH * NDET * 5 + bd] = ok ? (float)(cls + 1) : -1.0f;
      if (ok) scb[wi] = NEGV;
    }
    __syncthreads();
  }
}

extern "C" void kernel_launch(void* const* d_in, const int* in_sizes, int n_in,
                              void* d_out, int out_size, void* d_ws,
                              size_t ws_size, hipStream_t stream) {
  (void)in_sizes; (void)n_in; (void)out_size; (void)ws_size;
  const float* logits = (const float*)d_in[0];   // [B, N, C]
  const float* reg = (const float*)d_in[1];      // [B, N, C*4]
  const float* props = (const float*)d_in[2];    // [B, N, 4]
  float* out = (float*)d_out;

  float* sc = (float*)d_ws;                             // [B*80, N]      640 KB
  float* bx = sc + (size_t)BATCH * NFG * NPROP;         // [B*80, N, 4]   2.5 MB

  k1_decode_score<<<(BATCH * NPROP) / 128, 256, 0, stream>>>(logits, reg,
                                                             props, sc, bx);
  k2_nms<<<BATCH * NFG, 256, 0, stream>>>(sc, bx);
  k3_topd<<<BATCH, 1024, 0, stream>>>(sc, bx, out);
}